// DWTInverse_73624329388391
// MI455X (gfx1250) — compile-verified
//
#include <hip/hip_runtime.h>

typedef __attribute__((ext_vector_type(2))) float v2f;
typedef __attribute__((ext_vector_type(8))) float v8f;

__device__ __forceinline__ float lane_gather(int idx, float v) {
    // ds_bpermute_b32: dst[lane] = src[idx[lane]]; index in bytes
    int r = __builtin_amdgcn_ds_bpermute(idx << 2, __builtin_bit_cast(int, v));
    return __builtin_bit_cast(float, r);
}

__device__ __forceinline__ float ldnt(const float* p) {
    // streaming load: every input byte is read exactly once -> TH=NT
    return __builtin_nontemporal_load(p);
}

// Fused 2-level inverse Haar DWT.
//   yl  : (16,64,64,64)      yh1 : (16,64,3,64,64)
//   yh0 : (16,64,3,128,128)  out : (16,64,256,256)
// One wave handles coarse strip (i, j0..j0+15) of one (n,c) image:
//   stage-1 WMMA -> ll1 rows {2i,2i+1}, cols 2j0..2j0+31 (held in D regs)
//   4x stage-2 WMMA -> out rows 4i..4i+3, cols 4j0..4j0+63
__global__ __launch_bounds__(256) void idwt2_haar_fused_wmma(
    const float* __restrict__ yl,
    const float* __restrict__ yh0,
    const float* __restrict__ yh1,
    float* __restrict__ out)
{
    const int lane = threadIdx.x & 31;
    const int wv   = blockIdx.x * 8 + (threadIdx.x >> 5);

    const int jt  = wv & 3;            // j0 = 16*jt
    const int i   = (wv >> 2) & 63;
    const int cc  = (wv >> 8) & 63;
    const int n   = wv >> 14;
    const int img = n * 64 + cc;

    const int m  = lane & 15;
    const int hi = lane >> 4;          // 0: K=0,1 half   1: K=2,3 half

    // ---- A matrix: Haar synthesis butterfly, 16x4 f32 (2 VGPRs)
    // rows M=0,1,8,9 hold +-0.5 combos; others zero.
    // coef(M,K) = 0.5 * (K in {1,3} ? sgn_a : 1) * (K in {2,3} ? sgn_b : 1)
    const float valid = ((m & 6) == 0) ? 0.5f : 0.0f;   // m in {0,1,8,9}
    const float sa = (m & 1) ? -1.0f : 1.0f;            // a = m&1
    const float sb = (m & 8) ? -1.0f : 1.0f;            // b = (m>>3)&1
    v2f A;
    A.x = valid * (hi ? sb : 1.0f);    // K=0 (ll) or K=2 (hl)
    A.y = A.x * sa;                    // K=1 (lh) or K=3 (hh)

    const v8f zero = {};

    // ---- Stage 1: (yl, yh1) butterfly for 16 coarse pixels (i, j0+m)
    const int j0 = jt << 4;
    const float* pyl = yl  + (long)img * 4096       + i * 64 + j0 + m;          // ll
    const float* ph1 = yh1 + (long)img * 3 * 4096   + i * 64 + j0 + m;          // s=0 lh
    // B: V0 lanes0-15 = ll (K0), lanes16-31 = hl (K2); V1 = lh (K1) | hh (K3)
    v2f B1;
    B1.x = hi ? ldnt(ph1 + 4096)     : ldnt(pyl);
    B1.y = hi ? ldnt(ph1 + 2 * 4096) : ldnt(ph1);
    v8f d1 = __builtin_amdgcn_wmma_f32_16x16x4_f32(
        false, A, false, B1, (short)0, zero, false, false);
    // d1[0]: ll1[2i  ][2j0 + colIdx]   d1[1]: ll1[2i+1][2j0 + colIdx]
    // where colIdx = 2*(lane&15) + (lane>>4)  (even cols lanes0-15, odd 16-31)

    // ---- Stage 2: (ll1, yh0) butterfly, 4 WMMAs
    const float* ph0 = yh0 + (long)img * 3 * 16384;     // s-plane stride 16384
    float* outp = out + (long)img * 65536;              // 256x256
    const int colIdx = (m << 1) | hi;
    // bpermute index: B column lane n needs d1 lane ((n&1)<<4) + 8*h2 + (n>>1)
    const int bidxBase = ((lane & 1) << 4) + (m >> 1);

#pragma unroll
    for (int r = 0; r < 2; ++r) {
        const float llsrc = (r == 0) ? d1[0] : d1[1];
        const int p = 2 * i + r;                        // level-0 row
#pragma unroll
        for (int h2 = 0; h2 < 2; ++h2) {
            const int q0 = 2 * j0 + 16 * h2;            // level-0 col base
            // lanes 0-15: ll1[p][q0+lane] gathered from stage-1 D registers
            const float llv = lane_gather(bidxBase + 8 * h2, llsrc);
            const float* base = ph0 + p * 128 + q0;
            const float hlv = ldnt(base + 16384 + m);   // s=1, one 64B segment
            v2f B2;
            B2.x = hi ? hlv : llv;                      // K0=ll1 | K2=hl0
            B2.y = hi ? ldnt(base + 2 * 16384 + m)      // K3=hh0
                      : ldnt(base + m);                 // K1=lh0
            v8f d2 = __builtin_amdgcn_wmma_f32_16x16x4_f32(
                false, A, false, B2, (short)0, zero, false, false);
            // rows 2p, 2p+1; 32 lanes cover 128B contiguous (interleaved cols)
            float* ob = outp + (long)(2 * p) * 256 + 2 * q0 + colIdx;
            __builtin_nontemporal_store(d2[0], ob);       // write-once -> TH=NT
            __builtin_nontemporal_store(d2[1], ob + 256);
        }
    }
}

extern "C" void kernel_launch(void* const* d_in, const int* in_sizes, int n_in,
                              void* d_out, int out_size, void* d_ws, size_t ws_size,
                              hipStream_t stream) {
    (void)in_sizes; (void)n_in; (void)out_size; (void)d_ws; (void)ws_size;
    const float* yl  = (const float*)d_in[0];   // (16,64,64,64)
    const float* yh0 = (const float*)d_in[1];   // (16,64,3,128,128)
    const float* yh1 = (const float*)d_in[2];   // (16,64,3,64,64)
    // d_in[3] = filts: Haar filters are hard-coded into the WMMA A matrix.
    float* outp = (float*)d_out;                // (16,64,256,256)

    // 2^18 wave-tiles, 8 waves (256 threads) per block
    idwt2_haar_fused_wmma<<<dim3(32768), dim3(256), 0, stream>>>(yl, yh0, yh1, outp);
}